// EmbeddingModule_38113539784739
// MI455X (gfx1250) — compile-verified
//
#include <hip/hip_runtime.h>
#include <hip/hip_bf16.h>

// Embedding gather: out[row, :] = weight[x[row], :]
//   rows = B*S = 16384, DIM = 768 fp32 = 192 float4 per row.
//
// MI455X strategy (compile-only reasoning):
//  - Pure bandwidth op, 0 FLOPs -> no WMMA. Optimize the VMEM path.
//  - 147 MB weight table fits the 192 MB L2: weight loads keep the default
//    temporal (RT) hint so the table stays L2-resident; the 48 MB output is
//    written with non-temporal 128-bit stores (th:TH_STORE_NT) so the output
//    stream does not evict the table.
//  - 4 token rows per block, 192 threads (6 wave32): each lane issues FOUR
//    independent global_load_b128s before any s_wait_loadcnt, giving 4x the
//    in-flight memory requests per wave (in-wave MLP) on top of cross-wave
//    occupancy. The 4 block-uniform token ids sit at a 16B-aligned address
//    (4*blockIdx.x ints) and coalesce into a single s_load_b128.

typedef __attribute__((ext_vector_type(4))) float v4f;

#ifndef EMB_DIM
#define EMB_DIM 768
#endif
#define VEC_PER_ROW   (EMB_DIM / 4)   // 192 float4 per row
#define ROWS_PER_BLK  4

__global__ __launch_bounds__(VEC_PER_ROW) void
embedding_gather_kernel(const int* __restrict__ x,
                        const v4f* __restrict__ weight,   // [VOCAB, 192] float4
                        v4f* __restrict__ out)            // [rows,  192] float4
{
    const int row0 = blockIdx.x * ROWS_PER_BLK;  // first token row of block
    const int col  = threadIdx.x;                // float4 column in [0, 192)

    // Block-uniform, 16B-aligned -> single scalar s_load_b128.
    int tok[ROWS_PER_BLK];
#pragma unroll
    for (int r = 0; r < ROWS_PER_BLK; ++r)
        tok[r] = x[row0 + r];

    // Issue 4 independent 128-bit gathers before any wait (in-wave MLP).
    v4f v[ROWS_PER_BLK];
#pragma unroll
    for (int r = 0; r < ROWS_PER_BLK; ++r)
        v[r] = weight[(size_t)tok[r] * VEC_PER_ROW + col];

    // Stream outputs with non-temporal 128-bit stores (protect L2 residency
    // of the embedding table).
#pragma unroll
    for (int r = 0; r < ROWS_PER_BLK; ++r)
        __builtin_nontemporal_store(v[r], &out[(size_t)(row0 + r) * VEC_PER_ROW + col]);
}

extern "C" void kernel_launch(void* const* d_in, const int* in_sizes, int n_in,
                              void* d_out, int out_size, void* d_ws, size_t ws_size,
                              hipStream_t stream)
{
    const int*   x      = (const int*)d_in[0];      // [B*S] token ids (int32)
    const float* weight = (const float*)d_in[1];    // [VOCAB, 768] fp32

    const int rows = in_sizes[0];                   // B*S = 16384 (divisible by 4)

    dim3 grid(rows / ROWS_PER_BLK);                 // 4096 blocks
    dim3 block(VEC_PER_ROW);                        // 192 threads = 6 wave32

    embedding_gather_kernel<<<grid, block, 0, stream>>>(
        x, (const v4f*)weight, (v4f*)d_out);
}